// LinearRegion_17772574670988
// MI455X (gfx1250) — compile-verified
//
#include <hip/hip_runtime.h>

#define BATCH      4096
#define LENGTH     128
#define FEAT       4
#define RANK       64
#define NSITES     (LENGTH - 2)          /* 126 */
#define SITE_ELEMS (RANK * FEAT * RANK)  /* 16384 floats = 64KB */
#define ROWS_PER_WG 64
#define WAVES_PER_WG (ROWS_PER_WG / 16)  /* 4 */
#define THREADS    (WAVES_PER_WG * 32)   /* 128 */

typedef float v2f __attribute__((ext_vector_type(2)));
typedef float v8f __attribute__((ext_vector_type(8)));
typedef unsigned int v4u __attribute__((ext_vector_type(4)));
typedef int  v8i __attribute__((ext_vector_type(8)));
typedef int  v4i __attribute__((ext_vector_type(4)));

// Issue a TDM load of one site's core tensor (16384 f32, contiguous) into LDS.
// D# built per CDNA5 ISA ch.8: group0 = {flags, lds_addr, global_addr lo/hi+type},
// group1 = {mask/data_size, tensor_dim0, tile_dim0, ...}, 1-D tensor == tile.
// This toolchain exposes the 6-arg builtin: (u32x4, i32x8, i32x4, i32x4, i32x8, i32 cpol).
__device__ __forceinline__ void tdm_load_site(const float* gsrc, unsigned lds_byte_addr) {
  unsigned long long ga = (unsigned long long)(uintptr_t)gsrc;
  v4u g0;
  g0.x = 1u;                                        // count=1, user descriptor
  g0.y = lds_byte_addr;                             // lds_addr [63:32]
  g0.z = (unsigned)(ga & 0xffffffffull);            // global_addr [95:64]
  g0.w = (unsigned)((ga >> 32) & 0x1ffffffull)      // global_addr [120:96]
         | (2u << 30);                              // type=2 ("image")
  unsigned td0 = (unsigned)SITE_ELEMS;              // tensor length (elements)
  v8i g1;
  g1[0] = (int)(2u << 16);                          // workgroup_mask=0, data_size=2 (4B)
  g1[1] = (int)((td0 & 0xffffu) << 16);             // tensor_dim0[15:0] @ bits 63:48
  g1[2] = (int)(td0 >> 16);                         // tensor_dim0[31:16]; tensor_dim1=0
  g1[3] = (int)((td0 & 0xffffu) << 16);             // tile_dim0 = 16384 @ bits 127:112
  g1[4] = 0;                                        // tile_dim1=0, tile_dim2=0 (unused)
  g1[5] = (int)td0;                                 // tensor_dim0_stride lo (benign)
  g1[6] = 0;
  g1[7] = 0;
  v4i z4 = {0, 0, 0, 0};
  v8i z8 = {0, 0, 0, 0, 0, 0, 0, 0};
  __builtin_amdgcn_tensor_load_to_lds(g0, g1, z4, z4, z8, 0);
}

__global__ __launch_bounds__(THREADS)
void tt_chain_kernel(const float* __restrict__ input,   // [B, L, F]
                     const float* __restrict__ first,   // [F, R]
                     const float* __restrict__ mid,     // [S, R, F, R]
                     const float* __restrict__ last,    // [R, F]
                     float* __restrict__ out)           // [B, 1]
{
  extern __shared__ float smem[];
  float* cbuf0 = smem;                      // 16384 f32 (site core, ping)
  float* cbuf1 = smem + SITE_ELEMS;         // 16384 f32 (site core, pong)
  float* vbuf  = smem + 2 * SITE_ELEMS;     // ROWS_PER_WG * RANK f32 (state V)

  const int t      = threadIdx.x;
  const int wave   = t >> 5;
  const int lane   = t & 31;
  const int lane15 = lane & 15;
  const int half   = lane >> 4;             // 0: lanes 0-15, 1: lanes 16-31
  const int i0     = half * 2;              // feature pair handled by this half
  const int wgBase = blockIdx.x * ROWS_PER_WG;

  // ---- init: V0[b,r] = sum_i first[i,r] * x[b,0,i]  (rows are wave-private) ----
  {
    int rowl = t >> 1;                      // 0..63; row t>>1 belongs to wave t>>5
    int row  = wgBase + rowl;
    const float* xp = input + (size_t)row * (LENGTH * FEAT);
    float x0 = xp[0], x1 = xp[1], x2 = xp[2], x3 = xp[3];
    int c0 = (t & 1) * (RANK / 2);
#pragma unroll
    for (int r = 0; r < RANK / 2; ++r) {
      int rr = c0 + r;
      vbuf[rowl * RANK + rr] = x0 * first[rr] + x1 * first[RANK + rr] +
                               x2 * first[2 * RANK + rr] + x3 * first[3 * RANK + rr];
    }
  }

  // Prologue: DMA site 0 core into cbuf0 (one wave issues; EXEC-independent DMA).
  if (wave == 0) {
    tdm_load_site(mid, (unsigned)(uintptr_t)cbuf0);
  }

  const int vrowbase = (wave * 16 + lane15) * RANK;
  const float* xrow = input + (size_t)(wgBase + wave * 16 + lane15) * (LENGTH * FEAT);

  for (int s = 0; s < NSITES; ++s) {
    float* cb = (s & 1) ? cbuf1 : cbuf0;
    if (wave == 0) {
      __builtin_amdgcn_s_wait_tensorcnt(0);      // site s tile resident in LDS
    }
    __syncthreads();                             // tile visible; prev buffer free
    if (wave == 0 && (s + 1) < NSITES) {
      float* nb = (s & 1) ? cbuf0 : cbuf1;
      tdm_load_site(mid + (size_t)(s + 1) * SITE_ELEMS, (unsigned)(uintptr_t)nb);
    }

    // Per-lane feature scalars for this site: x[row, s+1, i0], x[row, s+1, i0+1]
    v2f xf = *(const v2f*)(xrow + (size_t)(s + 1) * FEAT + i0);

    v8f acc0 = {}, acc1 = {}, acc2 = {}, acc3 = {};
    const float* bbase = cb + i0 * RANK + lane15;

#pragma unroll 2
    for (int kk = 0; kk < RANK; ++kk) {
      // A'[b, 4*kk + i] = V[b,kk] * x[b,i]  -> matches 32-bit A 16x4 VGPR layout
      float aS = vbuf[vrowbase + kk];
      v2f a;  a.x = aS * xf.x;  a.y = aS * xf.y;
      // B = C_s[k, r] row-major: v0 holds K-row i0, v1 holds K-row i0+1
      const float* bp = bbase + kk * (FEAT * RANK);
      v2f b0, b1, b2, b3;
      b0.x = bp[0];   b0.y = bp[RANK];
      b1.x = bp[16];  b1.y = bp[RANK + 16];
      b2.x = bp[32];  b2.y = bp[RANK + 32];
      b3.x = bp[48];  b3.y = bp[RANK + 48];
      acc0 = __builtin_amdgcn_wmma_f32_16x16x4_f32(false, a, false, b0, (short)0, acc0, false, false);
      acc1 = __builtin_amdgcn_wmma_f32_16x16x4_f32(false, a, false, b1, (short)0, acc1, false, false);
      acc2 = __builtin_amdgcn_wmma_f32_16x16x4_f32(false, a, false, b2, (short)0, acc2, false, false);
      acc3 = __builtin_amdgcn_wmma_f32_16x16x4_f32(false, a, false, b3, (short)0, acc3, false, false);
    }

    // Write V_new from D layout (vgpr j -> row half*8+j, lane -> column).
    // Rows are wave-private and LDS ops are in-order per wave: reads above
    // retire before these stores, so no barrier is needed for V itself.
    int wrow = wave * 16 + half * 8;
#pragma unroll
    for (int j = 0; j < 8; ++j) {
      float* vp = vbuf + (wrow + j) * RANK + lane15;
      vp[0]  = acc0[j];
      vp[16] = acc1[j];
      vp[32] = acc2[j];
      vp[48] = acc3[j];
    }
  }

  __syncthreads();

  // ---- epilogue: out[b] = sum_l V[b,l] * (sum_i last[l,i] * x[b,L-1,i]) ----
  {
    int rowl = t >> 1;
    int row  = wgBase + rowl;
    const float* xp = input + ((size_t)row * LENGTH + (LENGTH - 1)) * FEAT;
    float x0 = xp[0], x1 = xp[1], x2 = xp[2], x3 = xp[3];
    int l0 = (t & 1) * (RANK / 2);
    float p = 0.f;
#pragma unroll 8
    for (int l = 0; l < RANK / 2; ++l) {
      int ll = l0 + l;
      const float* lc = last + ll * FEAT;
      float lv = x0 * lc[0] + x1 * lc[1] + x2 * lc[2] + x3 * lc[3];
      p += vbuf[rowl * RANK + ll] * lv;
    }
    p += __shfl_xor(p, 1, 32);
    if ((t & 1) == 0) out[row] = p;
  }
}

extern "C" void kernel_launch(void* const* d_in, const int* in_sizes, int n_in,
                              void* d_out, int out_size, void* d_ws, size_t ws_size,
                              hipStream_t stream) {
  const float* input = (const float*)d_in[0];   // [4096,128,4]
  const float* first = (const float*)d_in[1];   // [4,64]
  const float* mid   = (const float*)d_in[2];   // [126,64,4,64]
  const float* last  = (const float*)d_in[3];   // [64,4]
  float* out = (float*)d_out;                   // [4096,1]

  size_t smem = (size_t)(2 * SITE_ELEMS + ROWS_PER_WG * RANK) * sizeof(float); // 144KB
  (void)hipFuncSetAttribute((const void*)tt_chain_kernel,
                            hipFuncAttributeMaxDynamicSharedMemorySize, (int)smem);
  tt_chain_kernel<<<BATCH / ROWS_PER_WG, THREADS, smem, stream>>>(input, first, mid, last, out);
}